// gauss_map_24713241822141
// MI455X (gfx1250) — compile-verified
//
#include <hip/hip_runtime.h>
#include <hip/hip_bf16.h>

typedef __attribute__((ext_vector_type(2))) float v2f;
typedef __attribute__((ext_vector_type(8))) float v8f;

#define N_LI 8192
#define N_RA 2048
#define M_PTS 2048
#define ROWS_TOTAL ((N_LI + N_RA) * 9) /* 92160 */
#define GRID_MOD 513

// Each block: 8 waves, each wave handles one 16-row query tile.
// LDS: B matrix for all 2048 candidate points, 4 f32 each (32 KB).
__global__ __launch_bounds__(256) void nearest_dy_wmma_kernel(
    const int* __restrict__ li_bev, const int* __restrict__ ra_bev,
    const float* __restrict__ ra_pts, const int* __restrict__ ra_vox,
    float* __restrict__ out)
{
    __shared__ float sB[M_PTS * 4];
    __shared__ int   sCnt;

    const int tid = threadIdx.x;
    if (tid == 0) sCnt = 0;
    __syncthreads();

    // ---- Build B columns in LDS + dynamic count ----
    int local = 0;
#pragma unroll
    for (int i = 0; i < M_PTS / 256; ++i) {
        int j = i * 256 + tid;
        float vel = ra_pts[j * 5 + 4];
        float px = (float)ra_vox[j * 3 + 1];
        float py = (float)ra_vox[j * 3 + 2];
        bool dyn = fabsf(vel) > 0.1f;
        sB[j * 4 + 0] = dyn ? -2.0f * px : 0.0f;
        sB[j * 4 + 1] = dyn ? -2.0f * py : 0.0f;
        sB[j * 4 + 2] = dyn ? px * px + py * py : 1.0e30f;
        sB[j * 4 + 3] = dyn ? 1.0f : 0.0f;
        local += dyn ? 1 : 0;
    }
    atomicAdd(&sCnt, local);
    __syncthreads();
    const float mul = (sCnt > 1) ? 0.01f : 0.0f;

    // ---- Per-lane A setup (ISA 16x4 f32 A layout) ----
    const int lane = tid & 31;
    const int wave = tid >> 5;
    const int m    = lane & 15;   // row within tile / column within B tile
    const int half = lane >> 4;   // selects K{0,1} vs K{2,3}
    const int tileBase = (blockIdx.x * 8 + wave) * 16;
    const int r = tileBase + m;   // global output row (tiles never straddle li/ra)

    int c0, c1, s;
    if (r < N_LI * 9) {
        int n = r / 9; s = r - n * 9;
        c0 = li_bev[n * 2 + 0]; c1 = li_bev[n * 2 + 1];
    } else {
        int rr = r - N_LI * 9;
        int n = rr / 9; s = rr - n * 9;
        c0 = ra_bev[n * 2 + 0]; c1 = ra_bev[n * 2 + 1];
    }
    // INDEX_SHIFT: coord0 shift = f(s%3) in {0,-1,1}; coord1 shift = g(s/3) in {0,1,-1}
    const int SH0[3] = {0, -1, 1};
    const int SH1[3] = {0, 1, -1};
    int x = (c0 + SH0[s % 3] + GRID_MOD) % GRID_MOD;
    int y = (c1 + SH1[s / 3] + GRID_MOD) % GRID_MOD;
    float xf = (float)x, yf = (float)y;

    v2f a;
    if (half == 0) { a.x = xf;   a.y = yf; }             // K=0 (x), K=1 (y)
    else           { a.x = 1.0f; a.y = xf * xf + yf * yf; } // K=2 (1), K=3 (|a|^2)

    float dmin[8];
#pragma unroll
    for (int k = 0; k < 8; ++k) dmin[k] = 3.0e38f;

    const float2* sB2 = (const float2*)sB;  // sB2[j*2 + half] = (b[2h], b[2h+1])
    const v8f czero = (v8f)0.0f;

    // ---- 128 WMMA 16x16x4 f32 ops: D[i][j] = |q_i - p_j|^2, min-accumulate ----
#pragma unroll 4
    for (int jt = 0; jt < M_PTS / 16; ++jt) {
        float2 bb = sB2[(jt * 16 + m) * 2 + half];
        v2f b; b.x = bb.x; b.y = bb.y;
        v8f d = __builtin_amdgcn_wmma_f32_16x16x4_f32(
            /*neg_a=*/false, a, /*neg_b=*/false, b,
            /*c_mod=*/(short)0, czero, /*reuse_a=*/false, /*reuse_b=*/false);
#pragma unroll
        for (int k = 0; k < 8; ++k) dmin[k] = fminf(dmin[k], d[k]);
    }

    // ---- Min over N: reduce across each 16-lane half (N is across lanes) ----
#pragma unroll
    for (int k = 0; k < 8; ++k) {
        float v = dmin[k];
        v = fminf(v, __shfl_xor(v, 1, 32));
        v = fminf(v, __shfl_xor(v, 2, 32));
        v = fminf(v, __shfl_xor(v, 4, 32));
        v = fminf(v, __shfl_xor(v, 8, 32));
        dmin[k] = v;
    }

    // VGPR k holds row M=k (lanes 0-15) / M=k+8 (lanes 16-31); lane 0/16 write.
    if (m == 0) {
#pragma unroll
        for (int k = 0; k < 8; ++k)
            out[tileBase + half * 8 + k] = sqrtf(fmaxf(dmin[k], 0.0f)) * mul;
    }
}

extern "C" void kernel_launch(void* const* d_in, const int* in_sizes, int n_in,
                              void* d_out, int out_size, void* d_ws, size_t ws_size,
                              hipStream_t stream) {
    (void)in_sizes; (void)n_in; (void)out_size; (void)d_ws; (void)ws_size;
    const int*   li  = (const int*)d_in[0];   // li_bev_coors [8192,2] i32
    const int*   ra  = (const int*)d_in[1];   // ra_bev_coors [2048,2] i32
    const float* pts = (const float*)d_in[2]; // ra_pts [2048,5] f32
    const int*   vox = (const int*)d_in[3];   // ra_voxel_coords [2048,3] i32
    float* out = (float*)d_out;               // [92160] f32 (li_dis ++ ra_dis)

    dim3 grid(ROWS_TOTAL / 128); // 720 blocks * 8 waves * 16 rows = 92160
    nearest_dy_wmma_kernel<<<grid, 256, 0, stream>>>(li, ra, pts, vox, out);
}